// extract_blocks_41472204210548
// MI455X (gfx1250) — compile-verified
//
#include <hip/hip_runtime.h>

typedef __attribute__((ext_vector_type(2))) float v2f;
typedef __attribute__((ext_vector_type(8))) float v8f;

// Problem constants: B=32, C=128, D=128, K2=16, n=8
#define OUT_IM_OFF 524288u   // 32*8*128*16

// -----------------------------------------------------------------------------
// Kernel 1: for each (j, b, c) extract first column + trace of the j-th
// diagonal 16x16 block of H[b,c]. Writes out[..., 1:16] and packs inv_in
// (256 rows x 256 cols, row = j*32+b) into workspace for the WMMA GEMM.
// -----------------------------------------------------------------------------
__global__ __launch_bounds__(256) void extract_pack_kernel(
    const float* __restrict__ H_re, const float* __restrict__ H_im,
    float* __restrict__ out,
    float* __restrict__ inv_r, float* __restrict__ inv_i)
{
    int tid = blockIdx.x * 256 + threadIdx.x;   // 32768 total
    int c = tid & 127;
    int b = (tid >> 7) & 31;
    int j = tid >> 12;                          // 0..7

    // H[b][c] is a 128x128 row-major tile; block (j,j) starts at row j*16, col j*16.
    const size_t base = (((size_t)(b * 128 + c)) << 14) + (size_t)j * 2064; // j*16*128 + j*16
    const float* hr = H_re + base;
    const float* hi = H_im + base;

    float* outr = out + ((((size_t)(b * 8 + j)) * 128 + c) << 4);
    float* outi = outr + OUT_IM_OFF;

    float tr = 0.f, ti = 0.f;
#pragma unroll
    for (int k = 1; k < 16; ++k) {
        float vr = hr[k * 128];      // block[k][0]  (first column)
        float vi = hi[k * 128];
        tr += hr[k * 129];           // block[k][k]  (diagonal, same 64B segment)
        ti += hi[k * 129];
        outr[k] = vr;
        outi[k] = vi;
    }
    int r = j * 32 + b;              // GEMM row index
    inv_r[r * 256 + c]       = hr[0];   // block[0][0]
    inv_r[r * 256 + 128 + c] = tr;
    inv_i[r * 256 + c]       = hi[0];
    inv_i[r * 256 + 128 + c] = ti;
}

// -----------------------------------------------------------------------------
// Kernel 2: inv = inv_in (256x256) @ W^T (256x128) in full fp32 via
// V_WMMA_F32_16X16X4_F32. One wave per 16x16 output tile; 16x8 tiles per
// re/im plane -> 256 waves total. Result scattered into out[..., 0].
//
// Fragment layouts (CDNA5 ISA 7.12.2, 32-bit A 16x4 / B 4x16):
//   A: lane<16 holds M=lane, K=0(v0),1(v1); lane>=16 holds M=lane-16, K=2,3
//   B: lane<16 holds N=lane, K=0(v0),1(v1); lane>=16 holds N=lane-16, K=2,3
//   -> both fragments are a contiguous float2 load per lane.
//   D: VGPR g, lane l: M = g + (l>=16 ? 8 : 0), N = l%16.
// -----------------------------------------------------------------------------
__global__ __launch_bounds__(256) void wmma_gemm_kernel(
    const float* __restrict__ inv_r, const float* __restrict__ inv_i,
    const float* __restrict__ W_real, const float* __restrict__ W_imag,
    float* __restrict__ out)
{
    int lane = threadIdx.x & 31;
    int wid  = (blockIdx.x * 256 + threadIdx.x) >> 5;  // 0..255
    int im   = wid >> 7;                               // 0 = real, 1 = imag
    int t    = wid & 127;
    int tm   = t >> 3;                                 // row tile 0..15
    int tn   = t & 7;                                  // col tile 0..7

    const float* A = im ? inv_i : inv_r;               // 256 x 256
    const float* W = im ? W_imag : W_real;             // 128 x 256 (row-major)

    int half = lane >> 4;                              // 0 or 1 (K pair select)
    int l15  = lane & 15;
    const float* aptr = A + (size_t)(tm * 16 + l15) * 256 + half * 2;
    const float* bptr = W + (size_t)(tn * 16 + l15) * 256 + half * 2;

    v8f acc = {};
#pragma unroll 8
    for (int k0 = 0; k0 < 256; k0 += 4) {
        v2f a  = *(const v2f*)(aptr + k0);
        v2f bb = *(const v2f*)(bptr + k0);
        // 8 args: (neg_a, A, neg_b, B, c_mod, C, reuse_a, reuse_b)
        acc = __builtin_amdgcn_wmma_f32_16x16x4_f32(false, a, false, bb,
                                                    (short)0, acc, false, false);
    }

    // Scatter D into out[im][b][j][c][0]; row r = j*32+b, col = c.
    size_t imoff = (size_t)im * OUT_IM_OFF;
    int cc = tn * 16 + l15;
    int mbase = half * 8;
#pragma unroll
    for (int g = 0; g < 8; ++g) {
        int r  = tm * 16 + mbase + g;
        int jj = r >> 5;
        int bb2 = r & 31;
        out[imoff + ((size_t)((bb2 * 8 + jj) * 128 + cc) << 4)] = acc[g];
    }
}

extern "C" void kernel_launch(void* const* d_in, const int* in_sizes, int n_in,
                              void* d_out, int out_size, void* d_ws, size_t ws_size,
                              hipStream_t stream) {
    const float* H_re   = (const float*)d_in[0];
    const float* H_im   = (const float*)d_in[1];
    const float* W_real = (const float*)d_in[2];
    const float* W_imag = (const float*)d_in[3];
    float* out   = (float*)d_out;
    float* inv_r = (float*)d_ws;            // 256*256 f32 = 256 KB
    float* inv_i = inv_r + 256 * 256;       // another 256 KB

    // Pass 1: extraction + packing (bandwidth-bound part, ~64 MB effective traffic)
    extract_pack_kernel<<<128, 256, 0, stream>>>(H_re, H_im, out, inv_r, inv_i);
    // Pass 2: fp32 WMMA GEMM (256x256 @ 256x128, re+im) -> out channel 0
    wmma_gemm_kernel<<<32, 256, 0, stream>>>(inv_r, inv_i, W_real, W_imag, out);
}